// AttentionPairBias_18373870092376
// MI455X (gfx1250) — compile-verified
//
#include <hip/hip_runtime.h>
#include <hip/hip_bf16.h>
#include <math.h>

typedef __attribute__((ext_vector_type(2))) float v2f;
typedef __attribute__((ext_vector_type(8))) float v8f;

#define S_DIM 1024
#define CA 1024
#define CS 512
#define CZ 64
#define NH 16
#define HC 64

__device__ __forceinline__ float sigmoidf_(float x) {
    return 1.0f / (1.0f + __expf(-x));
}

// ---------------- LayerNorm over last dim (one block per row) ----------------
__global__ void ln_kernel(const float* __restrict__ x, const float* __restrict__ w,
                          float* __restrict__ y, int C) {
    __shared__ float red[256];
    __shared__ float red2[256];
    int row = blockIdx.x;
    const float4* xr = (const float4*)(x + (size_t)row * C);
    int C4 = C >> 2;
    float sum = 0.f, sq = 0.f;
    for (int c = threadIdx.x; c < C4; c += 256) {
        float4 v = xr[c];
        sum += v.x + v.y + v.z + v.w;
        sq  += v.x * v.x + v.y * v.y + v.z * v.z + v.w * v.w;
    }
    red[threadIdx.x] = sum; red2[threadIdx.x] = sq;
    __syncthreads();
    for (int off = 128; off > 0; off >>= 1) {
        if (threadIdx.x < off) {
            red[threadIdx.x]  += red[threadIdx.x + off];
            red2[threadIdx.x] += red2[threadIdx.x + off];
        }
        __syncthreads();
    }
    float m   = red[0] / C;
    float var = red2[0] / C - m * m;
    float inv = rsqrtf(var + 1e-5f);
    float4* yr = (float4*)(y + (size_t)row * C);
    for (int c = threadIdx.x; c < C4; c += 256) {
        float4 v = xr[c];
        v.x = (v.x - m) * inv; v.y = (v.y - m) * inv;
        v.z = (v.z - m) * inv; v.w = (v.w - m) * inv;
        if (w) {
            const float4 wv = ((const float4*)w)[c];
            v.x *= wv.x; v.y *= wv.y; v.z *= wv.z; v.w *= wv.w;
        }
        yr[c] = v;
    }
}

// ---------------- Generic batched NT GEMM via WMMA f32 16x16x4 ----------------
// C[m,n] = epilogue( scale * sum_k A[m,k] * B[n,k] )
// mode 0: val = acc*scale + (bias?bias[n]:0) + (addm?addm[m,n]:0)
// mode 1: val = sigmoid(acc*scale + bias[n]) * addm[m,n]
// grid: (N/64, M/64, batch); block 256 (8 waves); double-buffered LDS, b128 loads
__global__ void gemm_nt_kernel(const float* __restrict__ A, int lda, long long sA,
                               const float* __restrict__ B, int ldb, long long sB,
                               float* __restrict__ C, int ldc, long long sC,
                               const float* __restrict__ bias,
                               const float* __restrict__ addm, int ldadd, long long sAdd,
                               int K, float scale, int mode) {
    __shared__ float As[2][64][20];   // pitch 20 floats: 16B aligned, conflict-free frags
    __shared__ float Bs[2][64][20];
    const int z = blockIdx.z;
    A += (size_t)z * sA; B += (size_t)z * sB; C += (size_t)z * sC;
    if (addm) addm += (size_t)z * sAdd;
    const int tileM = blockIdx.y * 64;
    const int tileN = blockIdx.x * 64;
    const int tid  = threadIdx.x;
    const int wave = tid >> 5, lane = tid & 31;
    const int wm = wave >> 1, wn = wave & 1;     // 4 x 2 wave grid
    const int half = lane >> 4, l = lane & 15;
    const int lr = tid >> 2;            // 0..63 : tile row
    const int lc = (tid & 3) * 4;       // 0,4,8,12 : k-column (float4)

    const float* gA = A + (size_t)(tileM + lr) * lda + lc;
    const float* gB = B + (size_t)(tileN + lr) * ldb + lc;

    v8f acc0 = {}; v8f acc1 = {};
    const int nstages = K >> 4;

    // prologue: stage 0 into buffer 0
    {
        float4 ra = *(const float4*)gA;
        float4 rb = *(const float4*)gB;
        *(float4*)&As[0][lr][lc] = ra;
        *(float4*)&Bs[0][lr][lc] = rb;
    }
    __syncthreads();

    for (int s = 0; s < nstages; s++) {
        const int p = s & 1;
        float4 na = make_float4(0.f, 0.f, 0.f, 0.f);
        float4 nb = make_float4(0.f, 0.f, 0.f, 0.f);
        if (s + 1 < nstages) {
            na = *(const float4*)(gA + (s + 1) * 16);
            nb = *(const float4*)(gB + (s + 1) * 16);
        }
        // fragment loads for the whole stage, then back-to-back WMMAs
        v2f af[4], bf0[4], bf1[4];
        #pragma unroll
        for (int q = 0; q < 4; q++) {
            const int kc = q * 4 + half * 2;
            af[q].x  = As[p][wm * 16 + l][kc];
            af[q].y  = As[p][wm * 16 + l][kc + 1];
            bf0[q].x = Bs[p][wn * 32 + l][kc];
            bf0[q].y = Bs[p][wn * 32 + l][kc + 1];
            bf1[q].x = Bs[p][wn * 32 + 16 + l][kc];
            bf1[q].y = Bs[p][wn * 32 + 16 + l][kc + 1];
        }
        #pragma unroll
        for (int q = 0; q < 4; q++) {
            acc0 = __builtin_amdgcn_wmma_f32_16x16x4_f32(false, af[q], false, bf0[q], (short)0, acc0, false, false);
            acc1 = __builtin_amdgcn_wmma_f32_16x16x4_f32(false, af[q], false, bf1[q], (short)0, acc1, false, false);
        }
        if (s + 1 < nstages) {
            *(float4*)&As[p ^ 1][lr][lc] = na;
            *(float4*)&Bs[p ^ 1][lr][lc] = nb;
        }
        __syncthreads();
    }

    #pragma unroll
    for (int r = 0; r < 8; r++) {
        int m  = tileM + wm * 16 + half * 8 + r;
        int n0 = tileN + wn * 32 + l;
        int n1 = n0 + 16;
        float v0 = acc0[r] * scale;
        float v1 = acc1[r] * scale;
        if (bias) { v0 += bias[n0]; v1 += bias[n1]; }
        if (mode == 1) {
            v0 = sigmoidf_(v0) * addm[(size_t)m * ldadd + n0];
            v1 = sigmoidf_(v1) * addm[(size_t)m * ldadd + n1];
        } else if (addm) {
            v0 += addm[(size_t)m * ldadd + n0];
            v1 += addm[(size_t)m * ldadd + n1];
        }
        C[(size_t)m * ldc + n0] = v0;
        C[(size_t)m * ldc + n1] = v1;
    }
}

// ---------------- Batched TN GEMM: C[m,n] = sum_k A[k,m] * B[k,n] ----------------
__global__ void gemm_tn_kernel(const float* __restrict__ A, int lda, long long sA,
                               const float* __restrict__ B, int ldb, long long sB,
                               float* __restrict__ C, int ldc, long long sC,
                               int K) {
    __shared__ float As[2][16][72];   // pitch 72: 16B aligned, disjoint half-wave banks
    __shared__ float Bs[2][16][72];
    const int z = blockIdx.z;
    A += (size_t)z * sA; B += (size_t)z * sB; C += (size_t)z * sC;
    const int tileM = blockIdx.y * 64;
    const int tileN = blockIdx.x * 64;
    const int tid  = threadIdx.x;
    const int wave = tid >> 5, lane = tid & 31;
    const int wm = wave >> 1, wn = wave & 1;
    const int half = lane >> 4, l = lane & 15;
    const int lk = tid >> 4;            // 0..15 : k row
    const int lm = (tid & 15) * 4;      // 0..60 : m column (float4)

    const float* gA = A + (size_t)lk * lda + tileM + lm;
    const float* gB = B + (size_t)lk * ldb + tileN + lm;

    v8f acc0 = {}; v8f acc1 = {};
    const int nstages = K >> 4;

    {
        float4 ra = *(const float4*)gA;
        float4 rb = *(const float4*)gB;
        *(float4*)&As[0][lk][lm] = ra;
        *(float4*)&Bs[0][lk][lm] = rb;
    }
    __syncthreads();

    for (int s = 0; s < nstages; s++) {
        const int p = s & 1;
        float4 na = make_float4(0.f, 0.f, 0.f, 0.f);
        float4 nb = make_float4(0.f, 0.f, 0.f, 0.f);
        if (s + 1 < nstages) {
            na = *(const float4*)(gA + (size_t)(s + 1) * 16 * lda);
            nb = *(const float4*)(gB + (size_t)(s + 1) * 16 * ldb);
        }
        v2f af[4], bf0[4], bf1[4];
        #pragma unroll
        for (int q = 0; q < 4; q++) {
            const int kc = q * 4 + half * 2;
            af[q].x  = As[p][kc][wm * 16 + l];
            af[q].y  = As[p][kc + 1][wm * 16 + l];
            bf0[q].x = Bs[p][kc][wn * 32 + l];
            bf0[q].y = Bs[p][kc + 1][wn * 32 + l];
            bf1[q].x = Bs[p][kc][wn * 32 + 16 + l];
            bf1[q].y = Bs[p][kc + 1][wn * 32 + 16 + l];
        }
        #pragma unroll
        for (int q = 0; q < 4; q++) {
            acc0 = __builtin_amdgcn_wmma_f32_16x16x4_f32(false, af[q], false, bf0[q], (short)0, acc0, false, false);
            acc1 = __builtin_amdgcn_wmma_f32_16x16x4_f32(false, af[q], false, bf1[q], (short)0, acc1, false, false);
        }
        if (s + 1 < nstages) {
            *(float4*)&As[p ^ 1][lk][lm] = na;
            *(float4*)&Bs[p ^ 1][lk][lm] = nb;
        }
        __syncthreads();
    }

    #pragma unroll
    for (int r = 0; r < 8; r++) {
        int m  = tileM + wm * 16 + half * 8 + r;
        int n0 = tileN + wn * 32 + l;
        C[(size_t)m * ldc + n0]      = acc0[r];
        C[(size_t)m * ldc + n0 + 16] = acc1[r];
    }
}

// ---------------- Row softmax over last dim (1024), in place, b128 ----------------
__global__ void softmax_row_kernel(float* __restrict__ x) {
    __shared__ float red[256];
    float4* xr = (float4*)(x + (size_t)blockIdx.x * 1024);
    float4 v = xr[threadIdx.x];
    float mx = fmaxf(fmaxf(v.x, v.y), fmaxf(v.z, v.w));
    red[threadIdx.x] = mx; __syncthreads();
    for (int off = 128; off > 0; off >>= 1) {
        if (threadIdx.x < off) red[threadIdx.x] = fmaxf(red[threadIdx.x], red[threadIdx.x + off]);
        __syncthreads();
    }
    mx = red[0];
    __syncthreads();
    v.x = __expf(v.x - mx); v.y = __expf(v.y - mx);
    v.z = __expf(v.z - mx); v.w = __expf(v.w - mx);
    float s = v.x + v.y + v.z + v.w;
    red[threadIdx.x] = s; __syncthreads();
    for (int off = 128; off > 0; off >>= 1) {
        if (threadIdx.x < off) red[threadIdx.x] += red[threadIdx.x + off];
        __syncthreads();
    }
    float inv = 1.0f / red[0];
    v.x *= inv; v.y *= inv; v.z *= inv; v.w *= inv;
    xr[threadIdx.x] = v;
}

// ------------- z layernorm + 16-head bias projection (256 MB streaming) -------------
__global__ void zbias_kernel(const float* __restrict__ z,
                             const float* __restrict__ pnw, const float* __restrict__ pnb,
                             const float* __restrict__ bw,  const float* __restrict__ bb,
                             float* __restrict__ out) {
    __shared__ float bwS[NH * CZ];
    __shared__ float pwS[CZ];
    __shared__ float pbS[CZ];
    __shared__ float bbS[NH];
    int tid = threadIdx.x;
    for (int i = tid; i < NH * CZ; i += 256) bwS[i] = bw[i];
    if (tid < CZ) { pwS[tid] = pnw[tid]; pbS[tid] = pnb[tid]; }
    if (tid < NH) bbS[tid] = bb[tid];
    __syncthreads();
    size_t pos = (size_t)blockIdx.x * 256 + tid;
    const float4* zp = (const float4*)(z + pos * CZ);
    float4 zv[16];
    float sum = 0.f, sq = 0.f;
    #pragma unroll
    for (int i = 0; i < 16; i++) {
        float4 t = zp[i];
        zv[i] = t;
        sum += t.x + t.y + t.z + t.w;
        sq  += t.x * t.x + t.y * t.y + t.z * t.z + t.w * t.w;
    }
    float m   = sum * (1.0f / 64.0f);
    float var = sq * (1.0f / 64.0f) - m * m;
    float inv = rsqrtf(var + 1e-5f);
    float o[NH];
    #pragma unroll
    for (int h = 0; h < NH; h++) o[h] = bbS[h];
    #pragma unroll
    for (int i = 0; i < 16; i++) {
        float y0 = (zv[i].x - m) * inv * pwS[4 * i + 0] + pbS[4 * i + 0];
        float y1 = (zv[i].y - m) * inv * pwS[4 * i + 1] + pbS[4 * i + 1];
        float y2 = (zv[i].z - m) * inv * pwS[4 * i + 2] + pbS[4 * i + 2];
        float y3 = (zv[i].w - m) * inv * pwS[4 * i + 3] + pbS[4 * i + 3];
        #pragma unroll
        for (int h = 0; h < NH; h++) {
            o[h] += y0 * bwS[h * 64 + 4 * i + 0] + y1 * bwS[h * 64 + 4 * i + 1]
                  + y2 * bwS[h * 64 + 4 * i + 2] + y3 * bwS[h * 64 + 4 * i + 3];
        }
    }
    float4* op = (float4*)(out + pos * NH);
    op[0] = make_float4(o[0],  o[1],  o[2],  o[3]);
    op[1] = make_float4(o[4],  o[5],  o[6],  o[7]);
    op[2] = make_float4(o[8],  o[9],  o[10], o[11]);
    op[3] = make_float4(o[12], o[13], o[14], o[15]);
}

// ---------------- a1 = sigmoid(pb * a_n + pn), in place into pb (b128) ----------------
__global__ void adaln_gate_kernel(const float4* __restrict__ pb, const float4* __restrict__ pn,
                                  const float4* __restrict__ an, float4* __restrict__ a1) {
    int i = blockIdx.x * 256 + threadIdx.x;
    float4 p = pb[i], q = pn[i], n = an[i];
    float4 r;
    r.x = sigmoidf_(p.x * n.x + q.x);
    r.y = sigmoidf_(p.y * n.y + q.y);
    r.z = sigmoidf_(p.z * n.z + q.z);
    r.w = sigmoidf_(p.w * n.w + q.w);
    a1[i] = r;
}

// ---------------- go = sigmoid(g_raw) * o  (b128) ----------------
__global__ void gate_o_kernel(const float* __restrict__ kvg, const float4* __restrict__ obuf,
                              float4* __restrict__ go) {
    int q = blockIdx.x * 256 + threadIdx.x;   // float4 index over flat [h][j][c]
    int h  = q >> 14;
    int j  = (q >> 4) & 1023;
    int c4 = (q & 15) * 4;
    const float4 g = *(const float4*)(kvg + (size_t)h * 196608 + (size_t)j * 192 + 128 + c4);
    float4 o = obuf[q];
    float4 r;
    r.x = sigmoidf_(g.x) * o.x;
    r.y = sigmoidf_(g.y) * o.y;
    r.z = sigmoidf_(g.z) * o.z;
    r.w = sigmoidf_(g.w) * o.w;
    go[q] = r;
}

extern "C" void kernel_launch(void* const* d_in, const int* in_sizes, int n_in,
                              void* d_out, int out_size, void* d_ws, size_t ws_size,
                              hipStream_t stream) {
    const float* a       = (const float*)d_in[0];
    const float* z       = (const float*)d_in[1];
    const float* s       = (const float*)d_in[2];
    const float* sn_w    = (const float*)d_in[3];
    const float* pb_w    = (const float*)d_in[4];
    const float* pb_b    = (const float*)d_in[5];
    const float* pn_w    = (const float*)d_in[6];
    const float* pnorm_w = (const float*)d_in[7];
    const float* pnorm_b = (const float*)d_in[8];
    const float* q_w     = (const float*)d_in[9];
    const float* q_b     = (const float*)d_in[10];
    const float* kvg_w   = (const float*)d_in[11];
    const float* bias_w  = (const float*)d_in[12];
    const float* bias_b  = (const float*)d_in[13];
    const float* attn_w  = (const float*)d_in[14];
    const float* out_w   = (const float*)d_in[15];
    const float* out_b   = (const float*)d_in[16];
    float* out = (float*)d_out;
    float* ws  = (float*)d_ws;

    // workspace layout (float offsets)
    float* a_n    = ws + 0;          // 1M
    float* s_n    = ws + 1048576;    // 0.5M
    float* pb     = ws + 1572864;    // 1M  (becomes a1 in place)
    float* pn     = ws + 2621440;    // 1M
    float* qm     = ws + 3670016;    // 1M
    float* kvg    = ws + 4718592;    // 3M
    float* biasO  = ws + 7864320;    // 16M ([x0][x1][h] flat == reshaped [h][i][j])
    float* obuf   = biasO;           // alias: bias consumed before O written
    float* go     = biasO + 1048576;
    float* a2     = biasO + 2097152;
    float* scores = ws + 24641536;   // 16M ([h][i][j])

    // 1. layernorms
    ln_kernel<<<1024, 256, 0, stream>>>(a, nullptr, a_n, 1024);
    ln_kernel<<<1024, 256, 0, stream>>>(s, sn_w, s_n, 512);

    // 2. pb = s_n @ pb_w.T + pb_b ; pn = s_n @ pn_w.T ; a1 = sigmoid(pb*a_n + pn)
    gemm_nt_kernel<<<dim3(16, 16, 1), 256, 0, stream>>>(
        s_n, 512, 0, pb_w, 512, 0, pb, 1024, 0, pb_b, nullptr, 0, 0, 512, 1.0f, 0);
    gemm_nt_kernel<<<dim3(16, 16, 1), 256, 0, stream>>>(
        s_n, 512, 0, pn_w, 512, 0, pn, 1024, 0, nullptr, nullptr, 0, 0, 512, 1.0f, 0);
    adaln_gate_kernel<<<1024, 256, 0, stream>>>(
        (const float4*)pb, (const float4*)pn, (const float4*)a_n, (float4*)pb);
    float* a1 = pb;

    // 3. q = a1 @ q_w.T + q_b ; kvg = a1 @ kvg_w.T
    gemm_nt_kernel<<<dim3(16, 16, 1), 256, 0, stream>>>(
        a1, 1024, 0, q_w, 1024, 0, qm, 1024, 0, q_b, nullptr, 0, 0, 1024, 1.0f, 0);
    gemm_nt_kernel<<<dim3(48, 16, 1), 256, 0, stream>>>(
        a1, 1024, 0, kvg_w, 1024, 0, kvg, 3072, 0, nullptr, nullptr, 0, 0, 1024, 1.0f, 0);

    // 4. bias = ln(z) @ bias_w.T + bias_b  -> biasO [S*S, 16] flat
    zbias_kernel<<<4096, 256, 0, stream>>>(z, pnorm_w, pnorm_b, bias_w, bias_b, biasO);

    // 5. scores[h,i,j] = (K_h @ Q_h^T)/64 + bias_r[h,i,j]   (batched over heads)
    gemm_nt_kernel<<<dim3(16, 16, 16), 256, 0, stream>>>(
        kvg, 192, 196608LL,           // A = K_h rows (stride 192 inside kvg block)
        qm, 64, 65536LL,              // B = Q_h
        scores, 1024, 1048576LL,
        nullptr,
        biasO, 1024, 1048576LL,       // + reshaped bias
        64, 1.0f / 64.0f, 0);

    // 6. P = row-softmax(scores)
    softmax_row_kernel<<<16384, 256, 0, stream>>>(scores);

    // 7. O_h = P_h^T @ V_h   -> obuf [h][j][c]
    gemm_tn_kernel<<<dim3(1, 16, 16), 256, 0, stream>>>(
        scores, 1024, 1048576LL,
        kvg + 64, 192, 196608LL,      // V_h
        obuf, 64, 65536LL,
        1024);

    // 8. go = sigmoid(g) * O   (flat [h][j][c] == reshape to [S, ca])
    gate_o_kernel<<<1024, 256, 0, stream>>>(kvg, (const float4*)obuf, (float4*)go);

    // 9. a2 = go @ attn_w.T
    gemm_nt_kernel<<<dim3(16, 16, 1), 256, 0, stream>>>(
        go, 1024, 0, attn_w, 1024, 0, a2, 1024, 0, nullptr, nullptr, 0, 0, 1024, 1.0f, 0);

    // 10. out = sigmoid(s @ out_w.T + out_b) * a2
    gemm_nt_kernel<<<dim3(16, 16, 1), 256, 0, stream>>>(
        s, 512, 0, out_w, 512, 0, out, 1024, 0, out_b, a2, 1024, 0, 512, 1.0f, 1);
}